// CRF__32091995635920
// MI455X (gfx1250) — compile-verified
//
#include <hip/hip_runtime.h>
#include <hip/hip_bf16.h>
#include <math.h>

typedef float v2f __attribute__((ext_vector_type(2)));
typedef float v8f __attribute__((ext_vector_type(8)));

#define L_SEQ 512
#define B_SZ  64
#define D_SZ  1024
#define T_SZ  20
#define TPAD  32
#define M_SZ  (L_SEQ * B_SZ)

#define NEG_BIG (-3.0e38f)

// ---------------- Kernel 0: pack fc_w (20x1024) -> zero-padded (32x1024) ----------------
__global__ __launch_bounds__(256) void crf_pack_fcw(const float* __restrict__ fc_w,
                                                    float* __restrict__ fcw_pad) {
    int idx = blockIdx.x * 256 + threadIdx.x;          // idx = col*1024 + k
    int col = idx >> 10;
    fcw_pad[idx] = (col < T_SZ) ? fc_w[idx] : 0.0f;
}

// ---------------- Kernel 1: emissions GEMM via V_WMMA_F32_16X16X4_F32 ----------------
// One wave computes a 16(M) x 32(Tpad) tile: two v8f accumulators share the A fragment.
__global__ __launch_bounds__(256) void crf_emit_gemm(const float* __restrict__ wf,
                                                     const float* __restrict__ fcw_pad,
                                                     const float* __restrict__ fc_b,
                                                     float* __restrict__ emit) {
    const int lane    = threadIdx.x & 31;
    const int wave    = threadIdx.x >> 5;
    const int lane_lo = lane & 15;
    const int lane_hi = lane >> 4;                    // 0: K={0,1}; 1: K={2,3}
    const int m0      = (blockIdx.x * 8 + wave) * 16;

    const float* arow = wf      + (size_t)(m0 + lane_lo) * D_SZ + 2 * lane_hi;
    const float* b0p  = fcw_pad + (size_t)lane_lo * D_SZ        + 2 * lane_hi;
    const float* b1p  = b0p + 16 * D_SZ;

    v8f c0 = {};
    v8f c1 = {};
    for (int k0 = 0; k0 < D_SZ; k0 += 4) {
        v2f a  = *(const v2f*)(arow + k0);
        v2f b0 = *(const v2f*)(b0p  + k0);
        v2f b1 = *(const v2f*)(b1p  + k0);
        c0 = __builtin_amdgcn_wmma_f32_16x16x4_f32(false, a, false, b0, (short)0, c0, false, false);
        c1 = __builtin_amdgcn_wmma_f32_16x16x4_f32(false, a, false, b1, (short)0, c1, false, false);
    }

    const float bias0 = fc_b[lane_lo];
    const float bias1 = (16 + lane_lo) < T_SZ ? fc_b[16 + lane_lo] : 0.0f;
#pragma unroll
    for (int j = 0; j < 8; ++j) {
        int row = m0 + j + 8 * lane_hi;               // C/D layout: vgpr j -> M=j (+8 for hi half)
        emit[(size_t)row * TPAD + lane_lo]      = c0[j] + bias0;
        emit[(size_t)row * TPAD + 16 + lane_lo] = c1[j] + bias1;
    }
}

// ---------------- Kernel 2: per-batch CRF scan (forward lse + Viterbi + gold score) ----------------
// One wave per batch; lane j (< 20) owns tag j.
__global__ __launch_bounds__(32) void crf_scan(const int* __restrict__ words,
                                               const int* __restrict__ tags,
                                               const float* __restrict__ emit,
                                               const float* __restrict__ start_t,
                                               const float* __restrict__ trans,
                                               const float* __restrict__ end_t,
                                               int* __restrict__ hist,
                                               float* __restrict__ loss,
                                               int* __restrict__ last_out,
                                               const int* __restrict__ pad_idx_p) {
    __shared__ float s_trans[T_SZ * T_SZ];
    const int b = blockIdx.x;
    const int j = threadIdx.x;
    const int pad_idx = *pad_idx_p;

    for (int i = j; i < T_SZ * T_SZ; i += 32) s_trans[i] = trans[i];
    __syncthreads();

    // column j of the transition matrix, in registers
    float tc[T_SZ];
#pragma unroll
    for (int i = 0; i < T_SZ; ++i) tc[i] = (j < T_SZ) ? s_trans[i * T_SZ + j] : 0.0f;

    float e0     = emit[(size_t)b * TPAD + j];        // row l=0 -> m = b
    float alpha  = (j < T_SZ) ? (start_t[j] + e0) : NEG_BIG;
    float vscore = alpha;

    int   prev  = tags[b];
    float score = (j == 0) ? (start_t[prev] + emit[(size_t)b * TPAD + prev]) : 0.0f;

    for (int l = 1; l < L_SEQ; ++l) {
        const size_t row = (size_t)l * B_SZ + b;
        const float  e   = emit[row * TPAD + j];
        const bool   mb  = (words[l * B_SZ + b] != pad_idx);   // uniform across the wave

        float cand[T_SZ];
        float m = NEG_BIG, vmax = NEG_BIG;
        int   bi = 0;
#pragma unroll
        for (int i = 0; i < T_SZ; ++i) {
            float ai = __shfl(alpha,  i, 32);
            float vi = __shfl(vscore, i, 32);
            cand[i]  = ai + tc[i];
            m        = fmaxf(m, cand[i]);
            float v  = vi + tc[i];
            if (v > vmax) { vmax = v; bi = i; }                // strict > keeps first argmax
        }
        float s = 0.0f;
#pragma unroll
        for (int i = 0; i < T_SZ; ++i) s += __expf(cand[i] - m);
        float nxt = m + __logf(s) + e;

        if (mb) { alpha = nxt; vscore = vmax + e; }
        if (j < T_SZ) hist[(size_t)(l - 1) * (B_SZ * T_SZ) + b * T_SZ + j] = bi;

        if (j == 0) {
            int tg = tags[l * B_SZ + b];
            if (mb) {
                score += s_trans[prev * T_SZ + tg] + emit[row * TPAD + tg];
                prev = tg;
            }
        }
    }

    // log_z = logsumexp_j(alpha + end)
    float av = (j < T_SZ) ? (alpha + end_t[j]) : NEG_BIG;
    float amax = av;
    for (int off = 16; off > 0; off >>= 1) amax = fmaxf(amax, __shfl_xor(amax, off, 32));
    float asum = __expf(av - amax);
    for (int off = 16; off > 0; off >>= 1) asum += __shfl_xor(asum, off, 32);
    float log_z = amax + __logf(asum);

    // last = argmax_j(vscore + end), ties -> lowest index
    float vv = (j < T_SZ) ? (vscore + end_t[j]) : NEG_BIG;
    int idx = j;
    for (int off = 16; off > 0; off >>= 1) {
        float ov = __shfl_xor(vv, off, 32);
        int   oi = __shfl_xor(idx, off, 32);
        if (ov > vv || (ov == vv && oi < idx)) { vv = ov; idx = oi; }
    }

    if (j == 0) {
        score += end_t[prev];
        loss[b]     = log_z - score;      // per-batch -(score - log_z)
        last_out[b] = idx;
    }
}

// ---------------- Kernel 3: backtrack + path output + ordered loss reduction ----------------
__global__ __launch_bounds__(64) void crf_backtrack(const int* __restrict__ words,
                                                    const int* __restrict__ hist,
                                                    const float* __restrict__ loss,
                                                    const int* __restrict__ last_in,
                                                    float* __restrict__ out,
                                                    const int* __restrict__ pad_idx_p) {
    const int b = threadIdx.x;
    const int pad_idx = *pad_idx_p;

    int cur = last_in[b];
    out[(size_t)(L_SEQ - 1) * B_SZ + b] = (float)cur;
    for (int l = L_SEQ - 2; l >= 0; --l) {
        int prev = hist[(size_t)l * (B_SZ * T_SZ) + b * T_SZ + cur];
        if (words[(l + 1) * B_SZ + b] != pad_idx) cur = prev;
        out[(size_t)l * B_SZ + b] = (float)cur;
    }

    if (b == 0) {
        float s = 0.0f;
        for (int i = 0; i < B_SZ; ++i) s += loss[i];   // fixed order -> deterministic
        out[(size_t)L_SEQ * B_SZ] = s;
    }
}

extern "C" void kernel_launch(void* const* d_in, const int* in_sizes, int n_in,
                              void* d_out, int out_size, void* d_ws, size_t ws_size,
                              hipStream_t stream) {
    const int*   words   = (const int*)d_in[0];
    const float* wf      = (const float*)d_in[1];
    const int*   tags    = (const int*)d_in[2];
    const float* fc_w    = (const float*)d_in[3];
    const float* fc_b    = (const float*)d_in[4];
    const float* start_t = (const float*)d_in[5];
    const float* trans   = (const float*)d_in[6];
    const float* end_t   = (const float*)d_in[7];
    const int*   pad_p   = (const int*)d_in[8];
    float*       out     = (float*)d_out;

    // workspace carve-up
    float* ws_fcw  = (float*)d_ws;                                   // 32*1024 f32
    float* ws_emit = ws_fcw + 32 * D_SZ;                             // M*32 f32 (4 MB)
    int*   ws_hist = (int*)(ws_emit + (size_t)M_SZ * TPAD);          // (L-1)*B*T i32
    float* ws_loss = (float*)(ws_hist + (size_t)(L_SEQ - 1) * B_SZ * T_SZ); // B f32
    int*   ws_last = (int*)(ws_loss + B_SZ);                         // B i32

    crf_pack_fcw<<<(32 * D_SZ) / 256, 256, 0, stream>>>(fc_w, ws_fcw);
    crf_emit_gemm<<<M_SZ / (16 * 8), 256, 0, stream>>>(wf, ws_fcw, fc_b, ws_emit);
    crf_scan<<<B_SZ, 32, 0, stream>>>(words, tags, ws_emit, start_t, trans, end_t,
                                      ws_hist, ws_loss, ws_last, pad_p);
    crf_backtrack<<<1, B_SZ, 0, stream>>>(words, ws_hist, ws_loss, ws_last, out, pad_p);
}